// HeteroDCVRepresentationModule_53352083751272
// MI455X (gfx1250) — compile-verified
//
#include <hip/hip_runtime.h>
#include <math.h>

// ---------------- problem constants (match reference) ----------------
#define ND 100000
#define NI 500000
#define NA 500000
#define NDOP 3
#define ED 16
#define NS 5
#define NL 3
// NOTE: ND, NI, NA are all multiples of 16 -> every WMMA tile is full,
// no tail predication needed (keeps EXEC all-ones around v_wmma).

typedef __attribute__((ext_vector_type(2))) float v2f;
typedef __attribute__((ext_vector_type(8))) float v8f;

// ---------------- small device helpers ----------------
__device__ __forceinline__ float dev_silu(float x) { return x / (1.f + __expf(-x)); }
__device__ __forceinline__ float dev_lrelu(float x) { return x > 0.f ? x : 0.2f * x; }

__device__ __forceinline__ void atomicMaxF(float* addr, float val) {
  if (val >= 0.f) atomicMax((int*)addr, __float_as_int(val));
  else            atomicMin((unsigned int*)addr, __float_as_uint(val));
}

__device__ __forceinline__ float wave_sum(float v) {
  #pragma unroll
  for (int o = 16; o > 0; o >>= 1) v += __shfl_down(v, o, 32);
  return v;
}

// FiLM MLP: cond[NCND] -> 16 -> 16 -> 32 ; x = g*x + b
template <int NCND>
__device__ __forceinline__ void film_apply(float* x, const float* cond,
    const float* __restrict__ W0, const float* __restrict__ b0,
    const float* __restrict__ W1, const float* __restrict__ b1,
    const float* __restrict__ W2, const float* __restrict__ b2) {
  float h0[16], h1[16];
  #pragma unroll
  for (int o = 0; o < 16; o++) {
    float s = b0[o];
    #pragma unroll
    for (int i = 0; i < NCND; i++) s += cond[i] * W0[i * 16 + o];
    h0[o] = dev_silu(s);
  }
  #pragma unroll
  for (int o = 0; o < 16; o++) {
    float s = b1[o];
    #pragma unroll
    for (int i = 0; i < 16; i++) s += h0[i] * W1[i * 16 + o];
    h1[o] = dev_silu(s);
  }
  #pragma unroll
  for (int o = 0; o < 16; o++) {
    float g = b2[o], b = b2[16 + o];
    #pragma unroll
    for (int i = 0; i < 16; i++) {
      g += h1[i] * W2[i * 32 + o];
      b += h1[i] * W2[i * 32 + 16 + o];
    }
    x[o] = g * x[o] + b;
  }
}

// interaction block attrs for one pair -> ii[NS]
__device__ __forceinline__ void pair_ii(const float* __restrict__ geo,
    const float* __restrict__ concs, int i0, int i1,
    const float* __restrict__ sig, float* ii) {
  float ri0 = geo[2 * i0], ri1 = geo[2 * i0 + 1];
  float rj0 = geo[2 * i1], rj1 = geo[2 * i1 + 1];
  const float c43pi = 4.18879020478639098f; // 4/3*pi
  float vi = c43pi * (ri1 * ri1 * ri1 - ri0 * ri0 * ri0);
  float vj = c43pi * (rj1 * rj1 * rj1 - rj0 * rj0 * rj0);
  float dm = 0.5f * (ri0 + ri1) - 0.5f * (rj0 + rj1);
  float d2 = dm * dm;
  float vv = vi * vj * concs[i0] * concs[i1];
  #pragma unroll
  for (int s = 0; s < NS; s++) {
    float sg = sig[s];
    ii[s] = vv * __expf(-d2 / (2.f * sg * sg));
  }
}

// ---------------- generic fill ----------------
__global__ void k_fill(float* p, float v, int n) {
  int i = blockIdx.x * blockDim.x + threadIdx.x;
  if (i < n) p[i] = v;
}

// ---------------- dopant node features (pre-BN) + BN stats ----------------
__global__ void k_dopant(const int* __restrict__ types, const float* __restrict__ concs,
    const int* __restrict__ cidx, const float* __restrict__ radii,
    const int* __restrict__ cradii, const float* __restrict__ emb,
    const float* fd0, const float* fd1, const float* fd2, const float* fd3,
    const float* fd4, const float* fd5,
    const float* fc0, const float* fc1, const float* fc2, const float* fc3,
    const float* fc4, const float* fc5,
    float* __restrict__ geo, float* __restrict__ xd, float* __restrict__ bnsum, int n) {
  int i = blockIdx.x * blockDim.x + threadIdx.x;
  bool valid = i < n;
  float x[16];
  #pragma unroll
  for (int k = 0; k < 16; k++) x[k] = 0.f;
  if (valid) {
    int c = cidx[i];
    float r0 = radii[cradii[c * 2]];
    float r1 = radii[cradii[c * 2 + 1]];
    geo[2 * i] = r0; geo[2 * i + 1] = r1;
    int t = types[i];
    #pragma unroll
    for (int k = 0; k < 16; k++) x[k] = emb[t * 16 + k];
    float cond1[1] = { concs[i] };
    film_apply<1>(x, cond1, fd0, fd1, fd2, fd3, fd4, fd5);
    float cond2[2] = { r0, r1 };
    film_apply<2>(x, cond2, fc0, fc1, fc2, fc3, fc4, fc5);
    #pragma unroll
    for (int k = 0; k < 16; k++) xd[i * 16 + k] = x[k];
  }
  // BN stats (sum, sumsq) per feature
  int lane = threadIdx.x & 31;
  #pragma unroll
  for (int k = 0; k < 16; k++) {
    float s  = wave_sum(x[k]);
    float sq = wave_sum(x[k] * x[k]);
    if (lane == 0) { atomicAdd(bnsum + k, s); atomicAdd(bnsum + 16 + k, sq); }
  }
}

// ---------------- BN finalize: scale/shift from sums ----------------
__global__ void k_bn_finalize(const float* __restrict__ bnsum,
    const float* __restrict__ gamma, const float* __restrict__ beta,
    float* __restrict__ sc, int dim, float n) {
  int k = threadIdx.x;
  if (k < dim) {
    float mean = bnsum[k] / n;
    float var  = bnsum[dim + k] / n - mean * mean;
    float s = gamma[k] * rsqrtf(var + 1e-5f);
    sc[k] = s;
    sc[dim + k] = beta[k] - mean * s;
  }
}

__global__ void k_bn_apply(float* __restrict__ x, const float* __restrict__ sc, int n) {
  int i = blockIdx.x * blockDim.x + threadIdx.x;
  if (i < n * 16) {
    int k = i & 15;
    x[i] = x[i] * sc[k] + sc[16 + k];
  }
}

// ---------------- pair pass 1: BN stats of interaction attrs ----------------
__global__ void k_pair1(const int* __restrict__ idx, const float* __restrict__ geo,
    const float* __restrict__ concs, const float* __restrict__ sig,
    float* __restrict__ bnsum, int n) {
  int e = blockIdx.x * blockDim.x + threadIdx.x;
  float ii[NS];
  #pragma unroll
  for (int s = 0; s < NS; s++) ii[s] = 0.f;
  if (e < n) pair_ii(geo, concs, idx[2 * e], idx[2 * e + 1], sig, ii);
  int lane = threadIdx.x & 31;
  #pragma unroll
  for (int s = 0; s < NS; s++) {
    float a  = wave_sum(ii[s]);
    float sq = wave_sum(ii[s] * ii[s]);
    if (lane == 0) { atomicAdd(bnsum + s, a); atomicAdd(bnsum + NS + s, sq); }
  }
}

// ---------------- pair pass 2: BN + FiLM -> pair node features ----------------
__global__ void k_pair2(const int* __restrict__ types, const int* __restrict__ idx,
    const float* __restrict__ geo, const float* __restrict__ concs,
    const float* __restrict__ sig, const float* __restrict__ sc,
    const float* __restrict__ emb,
    const float* f0, const float* f1, const float* f2, const float* f3,
    const float* f4, const float* f5,
    float* __restrict__ out, int n) {
  int e = blockIdx.x * blockDim.x + threadIdx.x;
  if (e >= n) return;
  float ii[NS];
  pair_ii(geo, concs, idx[2 * e], idx[2 * e + 1], sig, ii);
  float cond[NS];
  #pragma unroll
  for (int s = 0; s < NS; s++) cond[s] = ii[s] * sc[s] + sc[NS + s];
  int t = types[e];
  float x[16];
  #pragma unroll
  for (int k = 0; k < 16; k++) x[k] = emb[t * 16 + k];
  film_apply<NS>(x, cond, f0, f1, f2, f3, f4, f5);
  #pragma unroll
  for (int k = 0; k < 16; k++) out[e * 16 + k] = x[k];
}

// ---------------- WMMA GEMM: Y[n x 16] = X[n x 16] @ W[16 x 16] + b ----------------
// One wave per 16-row tile; K=16 via 4 chained V_WMMA_F32_16X16X4_F32.
// Requires n % 16 == 0 (all node counts are), so no predication anywhere.
__global__ void k_gemm16_wmma(const float* __restrict__ X, const float* __restrict__ W,
    const float* __restrict__ bvec, float* __restrict__ Y, int n) {
  int gwave = (blockIdx.x * blockDim.x + threadIdx.x) >> 5;
  int tiles = n >> 4;
  if (gwave >= tiles) return;   // uniform per wave: EXEC stays all-ones
  int lane = threadIdx.x & 31;
  int half = lane >> 4;         // K-half selector for A/B fragments
  int lm   = lane & 15;         // row (A) / col (B,D)
  int row  = gwave * 16 + lm;
  const v2f* __restrict__ Xv = (const v2f*)X;   // kb is always even
  v8f c = {};
  #pragma unroll
  for (int kc = 0; kc < 4; kc++) {
    int kb = kc * 4 + half * 2;
    v2f a = Xv[row * 8 + (kb >> 1)];            // one b64 load
    v2f b;
    b.x = W[kb * 16 + lm];
    b.y = W[(kb + 1) * 16 + lm];
    c = __builtin_amdgcn_wmma_f32_16x16x4_f32(false, a, false, b, (short)0, c,
                                              false, false);
  }
  float bias = bvec[lm];
  #pragma unroll
  for (int v = 0; v < 8; v++) {
    int r = gwave * 16 + v + half * 8;
    Y[r * 16 + lm] = c[v] + bias;
  }
}

// ---------------- GAT edge pass 1: scores + segment max ----------------
__global__ void k_edge1(const int* __restrict__ ei, int E,
    const float* __restrict__ msg, const float* __restrict__ xr,
    const float* __restrict__ att, float* __restrict__ sedge, float* __restrict__ m) {
  int e = blockIdx.x * blockDim.x + threadIdx.x;
  if (e >= E) return;
  int s = ei[e], d = ei[E + e];
  const float4* __restrict__ mp = (const float4*)(msg + s * 16);
  const float4* __restrict__ xp = (const float4*)(xr + d * 16);
  float acc = 0.f;
  #pragma unroll
  for (int q = 0; q < 4; q++) {
    float4 mv = mp[q];
    float4 xv = xp[q];
    acc += dev_lrelu(mv.x + xv.x) * att[q * 4 + 0];
    acc += dev_lrelu(mv.y + xv.y) * att[q * 4 + 1];
    acc += dev_lrelu(mv.z + xv.z) * att[q * 4 + 2];
    acc += dev_lrelu(mv.w + xv.w) * att[q * 4 + 3];
  }
  sedge[e] = acc;
  atomicMaxF(m + d, acc);
}

// ---------------- GAT edge pass 2: exp weights + weighted aggregate ----------------
__global__ void k_edge2(const int* __restrict__ ei, int E,
    const float* __restrict__ msg, const float* __restrict__ sedge,
    const float* __restrict__ m, float* __restrict__ z, float* __restrict__ acc) {
  int e = blockIdx.x * blockDim.x + threadIdx.x;
  if (e >= E) return;
  int s = ei[e], d = ei[E + e];
  float ew = __expf(sedge[e] - m[d]);
  atomicAdd(z + d, ew);
  const float4* __restrict__ mp = (const float4*)(msg + s * 16);
  float* ap = acc + d * 16;
  #pragma unroll
  for (int q = 0; q < 4; q++) {
    float4 mv = mp[q];
    atomicAdd(ap + q * 4 + 0, ew * mv.x);
    atomicAdd(ap + q * 4 + 1, ew * mv.y);
    atomicAdd(ap + q * 4 + 2, ew * mv.z);
    atomicAdd(ap + q * 4 + 3, ew * mv.w);
  }
}

// ---------------- GAT finalize ----------------
// mode 0: out = silu(acc/z + bias)
// mode 1: out = acc/z + bias            (raw partial into ytmp)
// mode 2: out = silu(acc/z + bias + ytmp)
__global__ void k_gat_fin(const float* __restrict__ acc, const float* __restrict__ z,
    const float* __restrict__ bias, const float* __restrict__ ytmp,
    float* __restrict__ out, int n, int mode) {
  int i = blockIdx.x * blockDim.x + threadIdx.x;
  if (i >= n * 16) return;
  int d = i >> 4, k = i & 15;
  float zz = z[d];
  float y = (zz > 0.f ? acc[i] / zz : 0.f) + bias[k];
  if (mode == 2) y += ytmp[i];
  if (mode != 1) y = dev_silu(y);
  out[i] = y;
}

// ---------------- final row-sum reduction (n x 16 -> 16) ----------------
__global__ void k_rowsum(const float* __restrict__ x, float* __restrict__ out, int n) {
  __shared__ float sm[256];
  int col = threadIdx.x & 15;
  int rl  = threadIdx.x >> 4;  // 0..15
  float s = 0.f;
  for (int r = blockIdx.x * 16 + rl; r < n; r += gridDim.x * 16) s += x[r * 16 + col];
  sm[threadIdx.x] = s;
  __syncthreads();
  for (int step = 128; step >= 16; step >>= 1) {
    if ((int)threadIdx.x < step) sm[threadIdx.x] += sm[threadIdx.x + step];
    __syncthreads();
  }
  if (threadIdx.x < 16) atomicAdd(out + threadIdx.x, sm[threadIdx.x]);
}

// ---------------- host orchestration ----------------
static inline int cdiv(int a, int b) { return (a + b - 1) / b; }

extern "C" void kernel_launch(void* const* d_in, const int* in_sizes, int n_in,
                              void* d_out, int out_size, void* d_ws, size_t ws_size,
                              hipStream_t stream) {
  (void)in_sizes; (void)n_in; (void)out_size; (void)ws_size;
  // ---- inputs (pytree-flattened setup_inputs order) ----
  const int*   dopant_types = (const int*)d_in[0];
  const float* dopant_concs = (const float*)d_in[1];
  const int*   dopant_cidx  = (const int*)d_in[2];
  const int*   inter_types  = (const int*)d_in[3];
  const int*   inter_idx    = (const int*)d_in[4];
  const int*   intra_types  = (const int*)d_in[5];
  const int*   intra_idx    = (const int*)d_in[6];
  const int*   ei_d2i       = (const int*)d_in[7];
  const int*   ei_i2d       = (const int*)d_in[8];
  const int*   ei_d2a       = (const int*)d_in[9];
  const int*   ei_a2d       = (const int*)d_in[10];
  const float* radii        = (const float*)d_in[11];
  const int*   cradii       = (const int*)d_in[12];
  const float* emb_dopant   = (const float*)d_in[13];
  const float* emb_inter    = (const float*)d_in[14];
  const float* emb_intra    = (const float*)d_in[15];
  const float* sigmas       = (const float*)d_in[16];
  const float* fd[6]; for (int i = 0; i < 6; i++) fd[i] = (const float*)d_in[17 + i];
  const float* fc[6]; for (int i = 0; i < 6; i++) fc[i] = (const float*)d_in[23 + i];
  const float* fi[6]; for (int i = 0; i < 6; i++) fi[i] = (const float*)d_in[29 + i];
  const float* fa[6]; for (int i = 0; i < 6; i++) fa[i] = (const float*)d_in[35 + i];
  const float* bnd_g = (const float*)d_in[41]; const float* bnd_b = (const float*)d_in[42];
  const float* bni_g = (const float*)d_in[43]; const float* bni_b = (const float*)d_in[44];
  const float* bna_g = (const float*)d_in[45]; const float* bna_b = (const float*)d_in[46];
  const float* gWl   = (const float*)d_in[47];
  const float* gbl   = (const float*)d_in[48];
  const float* gWr   = (const float*)d_in[49];
  const float* gbr   = (const float*)d_in[50];
  const float* gatt  = (const float*)d_in[51];
  const float* gbias = (const float*)d_in[52];
  float* out = (float*)d_out;

  // ---- workspace layout (float offsets) ----
  float* w = (float*)d_ws;
  size_t off = 0;
  float* geo   = w + off; off += (size_t)2 * ND;
  float* xd_a  = w + off; off += (size_t)ND * 16;
  float* xd_b  = w + off; off += (size_t)ND * 16;
  float* xi_a  = w + off; off += (size_t)NI * 16;
  float* xi_b  = w + off; off += (size_t)NI * 16;
  float* xa_a  = w + off; off += (size_t)NA * 16;
  float* xa_b  = w + off; off += (size_t)NA * 16;
  float* msg   = w + off; off += (size_t)NI * 16;       // max(node counts)*16
  float* xr    = w + off; off += (size_t)NI * 16;
  float* sedge = w + off; off += (size_t)2 * NI;        // max edge count
  float* mbuf  = w + off; off += (size_t)NI;
  float* zbuf  = w + off; off += (size_t)NI;
  float* accb  = w + off; off += (size_t)NI * 16;
  float* ytmp  = w + off; off += (size_t)ND * 16;
  float* bnsum_d = w + off; off += 32;
  float* bnsum_i = w + off; off += 16;
  float* bnsum_a = w + off; off += 16;
  float* bnsc_d  = w + off; off += 32;
  float* bnsc_i  = w + off; off += 16;
  float* bnsc_a  = w + off; off += 16;

  const int B = 256;
  const float NEG_INF = -__builtin_inff();

  // ---- phase A: node features ----
  k_fill<<<1, 128, 0, stream>>>(bnsum_d, 0.f, 32 + 16 + 16);  // contiguous sums
  k_dopant<<<cdiv(ND, B), B, 0, stream>>>(dopant_types, dopant_concs, dopant_cidx,
      radii, cradii, emb_dopant,
      fd[0], fd[1], fd[2], fd[3], fd[4], fd[5],
      fc[0], fc[1], fc[2], fc[3], fc[4], fc[5],
      geo, xd_a, bnsum_d, ND);
  k_bn_finalize<<<1, 32, 0, stream>>>(bnsum_d, bnd_g, bnd_b, bnsc_d, 16, (float)ND);
  k_bn_apply<<<cdiv(ND * 16, B), B, 0, stream>>>(xd_a, bnsc_d, ND);

  k_pair1<<<cdiv(NI, B), B, 0, stream>>>(inter_idx, geo, dopant_concs, sigmas, bnsum_i, NI);
  k_pair1<<<cdiv(NA, B), B, 0, stream>>>(intra_idx, geo, dopant_concs, sigmas, bnsum_a, NA);
  k_bn_finalize<<<1, 32, 0, stream>>>(bnsum_i, bni_g, bni_b, bnsc_i, NS, (float)NI);
  k_bn_finalize<<<1, 32, 0, stream>>>(bnsum_a, bna_g, bna_b, bnsc_a, NS, (float)NA);
  k_pair2<<<cdiv(NI, B), B, 0, stream>>>(inter_types, inter_idx, geo, dopant_concs,
      sigmas, bnsc_i, emb_inter, fi[0], fi[1], fi[2], fi[3], fi[4], fi[5], xi_a, NI);
  k_pair2<<<cdiv(NA, B), B, 0, stream>>>(intra_types, intra_idx, geo, dopant_concs,
      sigmas, bnsc_a, emb_intra, fa[0], fa[1], fa[2], fa[3], fa[4], fa[5], xa_a, NA);

  // ---- phase B: GAT layers ----
  float* xd_cur = xd_a; float* xd_nxt = xd_b;
  float* xi_cur = xi_a; float* xi_nxt = xi_b;
  float* xa_cur = xa_a; float* xa_nxt = xa_b;

  auto run_gat = [&](const float* xs, int n_src, const float* xdn, int n_dst,
                     const int* ei, int E, int l, int t, float* outbuf, int mode) {
    const float* Wl = gWl + (size_t)(l * 4 + t) * 256;
    const float* bl = gbl + (size_t)(l * 4 + t) * 16;
    const float* Wr = gWr + (size_t)(l * 4 + t) * 256;
    const float* br = gbr + (size_t)(l * 4 + t) * 16;
    const float* at = gatt + (size_t)(l * 4 + t) * 16;
    const float* bs = gbias + (size_t)(l * 4 + t) * 16;
    int tiles_s = n_src >> 4, tiles_d = n_dst >> 4;   // n always multiple of 16
    k_gemm16_wmma<<<cdiv(tiles_s, 8), 256, 0, stream>>>(xs, Wl, bl, msg, n_src);
    k_gemm16_wmma<<<cdiv(tiles_d, 8), 256, 0, stream>>>(xdn, Wr, br, xr, n_dst);
    k_fill<<<cdiv(n_dst, B), B, 0, stream>>>(mbuf, NEG_INF, n_dst);
    k_fill<<<cdiv(n_dst, B), B, 0, stream>>>(zbuf, 0.f, n_dst);
    k_fill<<<cdiv(n_dst * 16, B), B, 0, stream>>>(accb, 0.f, n_dst * 16);
    k_edge1<<<cdiv(E, B), B, 0, stream>>>(ei, E, msg, xr, at, sedge, mbuf);
    k_edge2<<<cdiv(E, B), B, 0, stream>>>(ei, E, msg, sedge, mbuf, zbuf, accb);
    k_gat_fin<<<cdiv(n_dst * 16, B), B, 0, stream>>>(accb, zbuf, bs, ytmp, outbuf,
                                                     n_dst, mode);
  };

  for (int l = 0; l < NL; l++) {
    run_gat(xd_cur, ND, xi_cur, NI, ei_d2i, 2 * NI, l, 0, xi_nxt, 0); // y_i
    run_gat(xi_cur, NI, xd_cur, ND, ei_i2d, 2 * NI, l, 1, ytmp, 1);   // partial y_d
    run_gat(xa_cur, NA, xd_cur, ND, ei_a2d, 2 * NA, l, 3, xd_nxt, 2); // y_d += , silu
    run_gat(xd_cur, ND, xa_cur, NA, ei_d2a, 2 * NA, l, 2, xa_nxt, 0); // y_a
    float* t;
    t = xd_cur; xd_cur = xd_nxt; xd_nxt = t;
    t = xi_cur; xi_cur = xi_nxt; xi_nxt = t;
    t = xa_cur; xa_cur = xa_nxt; xa_nxt = t;
  }

  // ---- final reduction ----
  k_fill<<<1, 32, 0, stream>>>(out, 0.f, 16);
  k_rowsum<<<64, 256, 0, stream>>>(xd_cur, out, ND);
}